// VectorQuantizer2_26551487824077
// MI455X (gfx1250) — compile-verified
//
#include <hip/hip_runtime.h>

typedef float v2f __attribute__((ext_vector_type(2)));
typedef float v8f __attribute__((ext_vector_type(8)));

#define DIMS        256
#define CHUNK       16          // codes per WMMA tile
#define ROWS_PB     128         // rows per block (8 waves x 16-row tiles)
#define THREADS     256
#define NWAVES      8
#define LDS_STRIDE  260         // 256 + 4 pad -> conflict-free 4-dword-stride access
#define WREGS       4           // float4 staging regs per thread (16*256/4/256)

// ---------------------------------------------------------------------------
// Kernel 0: wnorm[k] = sum_d W[k,d]^2
// ---------------------------------------------------------------------------
__global__ __launch_bounds__(THREADS) void vq_wnorm_kernel(
    const float* __restrict__ w, float* __restrict__ wnorm) {
  __shared__ float part[NWAVES];
  const int k = blockIdx.x;
  const int tid = threadIdx.x;
  float v = w[(size_t)k * DIMS + tid];
  v *= v;
  #pragma unroll
  for (int off = 16; off >= 1; off >>= 1) v += __shfl_xor(v, off, 32);
  if ((tid & 31) == 0) part[tid >> 5] = v;
  __syncthreads();
  if (tid == 0) {
    float s = 0.f;
    #pragma unroll
    for (int i = 0; i < NWAVES; ++i) s += part[i];
    wnorm[k] = s;
  }
}

// ---------------------------------------------------------------------------
// Kernel 1: main — WMMA f32 scores, double-buffered W staging, running argmin,
//           gather, per-block SSE
// ---------------------------------------------------------------------------
__global__ __launch_bounds__(THREADS) void vq_main_kernel(
    const float* __restrict__ x, const float* __restrict__ w,
    const float* __restrict__ wnorm, float* __restrict__ outq,
    float* __restrict__ outidx, float* __restrict__ blk_sse, int kchunks) {
  __shared__ float Xs[ROWS_PB * LDS_STRIDE];       // ~130 KB
  __shared__ float Ws[2][CHUNK * LDS_STRIDE];      // 2 x ~16.6 KB double buffer
  __shared__ int   idxs[ROWS_PB];
  __shared__ float wsum[NWAVES];

  const int tid  = threadIdx.x;
  const int lane = tid & 31;
  const int wave = tid >> 5;
  const int half = lane >> 4;     // 0: lanes 0-15, 1: lanes 16-31
  const int l16  = lane & 15;
  const size_t rowBase = (size_t)blockIdx.x * ROWS_PB;

  // Per-thread staging coordinates for the 16x256 W chunk (4 float4 each)
  int scc[WREGS], scol[WREGS];
  #pragma unroll
  for (int t = 0; t < WREGS; ++t) {
    const int e = (tid + t * THREADS) * 4;
    scc[t]  = e >> 8;    // local code row 0..15
    scol[t] = e & 255;   // column
  }

  // Stage X tile: 128 rows x 256 f32, padded stride
  for (int i = tid; i < ROWS_PB * DIMS / 4; i += THREADS) {
    const int e = i * 4;
    const int r = e >> 8;
    const int c = e & 255;
    *(float4*)&Xs[r * LDS_STRIDE + c] = *(const float4*)(x + rowBase * DIMS + e);
  }

  // Prologue: stage W chunk 0 into buffer 0
  float4 wreg[WREGS];
  #pragma unroll
  for (int t = 0; t < WREGS; ++t)
    wreg[t] = *(const float4*)(w + (size_t)scc[t] * DIMS + scol[t]);
  #pragma unroll
  for (int t = 0; t < WREGS; ++t)
    *(float4*)&Ws[0][scc[t] * LDS_STRIDE + scol[t]] = wreg[t];
  __syncthreads();

  float bestV[8];
  int   bestI[8];
  #pragma unroll
  for (int j = 0; j < 8; ++j) { bestV[j] = 3.4e38f; bestI[j] = 0; }

  // A frag (16x4 f32): lane l16 = row, half selects K pair {0,1}/{2,3}
  const float* aBase = &Xs[(wave * CHUNK + l16) * LDS_STRIDE + (half << 1)];
  const int bOff = l16 * LDS_STRIDE + (half << 1);

  for (int c = 0; c < kchunks; ++c) {
    const int buf = c & 1;

    // Issue global loads for chunk c+1 now; loadcnt stays pending behind the
    // 64-WMMA chain below (latency hiding).
    if (c + 1 < kchunks) {
      #pragma unroll
      for (int t = 0; t < WREGS; ++t)
        wreg[t] = *(const float4*)(
            w + ((size_t)((c + 1) * CHUNK + scc[t])) * DIMS + scol[t]);
    }

    const float wn = wnorm[c * CHUNK + l16];
    const float* bBase = &Ws[buf][bOff];

    v8f acc = {0.f, 0.f, 0.f, 0.f, 0.f, 0.f, 0.f, 0.f};
    #pragma unroll
    for (int kk = 0; kk < DIMS; kk += 4) {
      v2f a = *(const v2f*)(aBase + kk);
      v2f b = *(const v2f*)(bBase + kk);
      acc = __builtin_amdgcn_wmma_f32_16x16x4_f32(
          false, a, false, b, (short)0, acc, false, false);
    }

    // acc[j]: row = wave*16 + j + 8*half, code = c*16 + l16
    const int code = c * CHUNK + l16;
    #pragma unroll
    for (int j = 0; j < 8; ++j) {
      const float v = wn - 2.0f * acc[j];
      if (v < bestV[j]) { bestV[j] = v; bestI[j] = code; }   // earlier code wins ties
    }

    // Drain staged chunk c+1 into the alternate buffer. The barrier below is
    // the only one per iteration: it separates iteration c's reads of Ws[buf]
    // from iteration c+1's writes to Ws[buf] two iterations later.
    if (c + 1 < kchunks) {
      #pragma unroll
      for (int t = 0; t < WREGS; ++t)
        *(float4*)&Ws[buf ^ 1][scc[t] * LDS_STRIDE + scol[t]] = wreg[t];
    }
    __syncthreads();
  }

  // Cross-lane argmin over the 16 codes held per half-wave (offsets stay in-half)
  #pragma unroll
  for (int j = 0; j < 8; ++j) {
    float v = bestV[j];
    int idx = bestI[j];
    #pragma unroll
    for (int off = 8; off >= 1; off >>= 1) {
      const float ov = __shfl_xor(v, off, 32);
      const int   oi = __shfl_xor(idx, off, 32);
      if (ov < v || (ov == v && oi < idx)) { v = ov; idx = oi; }
    }
    if (l16 == 0) idxs[wave * CHUNK + j + (half << 3)] = idx;
  }
  __syncthreads();

  // Gather quantized rows, write out, accumulate SSE (X tile still in LDS)
  float sse = 0.f;
  for (int r = 0; r < ROWS_PB; ++r) {
    const int code = idxs[r];
    const float q  = w[(size_t)code * DIMS + tid];
    const float xv = Xs[r * LDS_STRIDE + tid];
    const float d  = q - xv;
    sse += d * d;
    outq[(rowBase + r) * DIMS + tid] = q;
  }
  if (tid < ROWS_PB) outidx[rowBase + tid] = (float)idxs[tid];

  #pragma unroll
  for (int off = 16; off >= 1; off >>= 1) sse += __shfl_xor(sse, off, 32);
  if (lane == 0) wsum[wave] = sse;
  __syncthreads();
  if (tid == 0) {
    float s = 0.f;
    #pragma unroll
    for (int i = 0; i < NWAVES; ++i) s += wsum[i];
    blk_sse[blockIdx.x] = s;
  }
}

// ---------------------------------------------------------------------------
// Kernel 2: deterministic loss reduction: 1.25 * SSE / (N*D)
// ---------------------------------------------------------------------------
__global__ void vq_loss_kernel(const float* __restrict__ blk_sse, int nblocks,
                               float* __restrict__ loss, float inv_count) {
  float s = 0.f;
  for (int i = 0; i < nblocks; ++i) s += blk_sse[i];
  *loss = 1.25f * s * inv_count;
}

// ---------------------------------------------------------------------------
extern "C" void kernel_launch(void* const* d_in, const int* in_sizes, int n_in,
                              void* d_out, int out_size, void* d_ws, size_t ws_size,
                              hipStream_t stream) {
  const float* x = (const float*)d_in[0];       // [16,4096,256] f32
  const float* w = (const float*)d_in[1];       // [1024,256]    f32

  const int nrows   = in_sizes[0] / DIMS;       // 65536
  const int K       = in_sizes[1] / DIMS;       // 1024
  const int nblocks = nrows / ROWS_PB;          // 512

  float* outq   = (float*)d_out;                // N*D quantized (== quantized_st fwd)
  float* loss   = outq + (size_t)nrows * DIMS;  // 1 scalar
  float* outidx = loss + 1;                     // N indices (as float)

  float* blk_sse = (float*)d_ws;                // [nblocks]
  float* wnorm   = blk_sse + nblocks;           // [K]

  vq_wnorm_kernel<<<K, THREADS, 0, stream>>>(w, wnorm);
  vq_main_kernel<<<nblocks, THREADS, 0, stream>>>(x, w, wnorm, outq, outidx,
                                                  blk_sse, K / CHUNK);
  vq_loss_kernel<<<1, 1, 0, stream>>>(blk_sse, nblocks, loss,
                                      1.0f / ((float)nrows * (float)DIMS));
}